// GATN_12910671692310
// MI455X (gfx1250) — compile-verified
//
#include <hip/hip_runtime.h>
#include <math.h>

#define NNODES 20000
#define NEG_SLOPE 0.2f
#define F_IN 50
#define KP1 52   // F_IN padded to multiple of 4
#define H1 4
#define C1 64
#define F1 256   // H1*C1
#define NCLS 121
#define LD2 128  // padded row stride for layer-2 xh / transposed W2

typedef __attribute__((ext_vector_type(2))) float v2f;
typedef __attribute__((ext_vector_type(8))) float v8f;

__global__ void k_fill(float* __restrict__ p, float v, int n) {
  int i = blockIdx.x * blockDim.x + threadIdx.x;
  if (i < n) p[i] = v;
}

// dst [R x Cout] = zero-padded copy of src [R x Cin], Cout >= Cin
__global__ void k_pad(const float* __restrict__ src, float* __restrict__ dst,
                      int R, int Cin, int Cout) {
  int i = blockIdx.x * blockDim.x + threadIdx.x;
  int total = R * Cout;
  if (i >= total) return;
  int r = i / Cout, c = i % Cout;
  dst[i] = (c < Cin) ? src[(size_t)r * Cin + c] : 0.0f;
}

// Bt [Npad x Kpad] = transpose of B [K x N], zero-padded
__global__ void k_padT(const float* __restrict__ B, float* __restrict__ Bt,
                       int K, int N, int Kpad, int Npad) {
  int i = blockIdx.x * blockDim.x + threadIdx.x;
  int total = Npad * Kpad;
  if (i >= total) return;
  int n = i / Kpad, k = i % Kpad;
  Bt[i] = (n < N && k < K) ? B[(size_t)k * N + n] : 0.0f;
}

// C[16x16 tile] = A[M x K] (row-major, K%4==0) * Bt[N x K] (transposed, K-padded).
// Both fragments are single 8B-aligned b64 loads at immediate offsets.
// 128-thread block = 4 waves; wave w owns N-tile (blockIdx.y*4 + w).
template <int K>
__launch_bounds__(128)
__global__ void k_gemm_wmma(const float* __restrict__ A, const float* __restrict__ Bt,
                            float* __restrict__ C, int ldc) {
  const int lane = threadIdx.x & 31;
  const int wv   = threadIdx.x >> 5;
  const int half = lane >> 4;     // 0: lanes 0-15, 1: lanes 16-31
  const int l16  = lane & 15;
  const int m0 = blockIdx.x * 16;
  const int n0 = (blockIdx.y * 4 + wv) * 16;
  // A frag: lanes 0-15 hold K=k0,k0+1 of row m0+l16; lanes 16-31 hold K=k0+2,k0+3
  const float* __restrict__ arow = A  + (size_t)(m0 + l16) * K + half * 2;
  // B frag: same K split, column n0+l16 of B == row n0+l16 of Bt
  const float* __restrict__ brow = Bt + (size_t)(n0 + l16) * K + half * 2;
  v8f acc = {};
#pragma unroll 4
  for (int k0 = 0; k0 < K; k0 += 4) {
    v2f a = *(const v2f*)(arow + k0);
    v2f b = *(const v2f*)(brow + k0);
    acc = __builtin_amdgcn_wmma_f32_16x16x4_f32(
        /*neg_a=*/false, a, /*neg_b=*/false, b,
        /*c_mod=*/(short)0, acc, /*reuse_a=*/false, /*reuse_b=*/false);
  }
  // C/D layout: VGPR v holds M = m0 + 8*half + v, N = n0 + l16
  float* crow = C + (size_t)(m0 + half * 8) * ldc + n0 + l16;
#pragma unroll
  for (int v = 0; v < 8; ++v) crow[(size_t)v * ldc] = acc[v];
}

// a[n*H+h] = dot(xh[n, h*C : h*C+C], att[h, :])
__global__ void k_att(const float* __restrict__ xh, const float* __restrict__ att,
                      float* __restrict__ a, int NH, int H, int C, int ldx) {
  int idx = blockIdx.x * blockDim.x + threadIdx.x;
  if (idx >= NH) return;
  int nn = idx / H, h = idx % H;
  const float* __restrict__ row = xh + (size_t)nn * ldx + h * C;
  const float* __restrict__ w = att + h * C;
  float s = 0.0f;
  for (int c = 0; c < C; ++c) s += row[c] * w[c];
  a[idx] = s;
}

__device__ __forceinline__ void edge_nodes(const int* __restrict__ ei, int E, int e,
                                           int& s, int& d) {
  if (e < E) { s = ei[e]; d = ei[E + e]; }   // edge_index row 0 = src, row 1 = dst
  else       { s = e - E; d = e - E; }       // appended self loops
}

__device__ __forceinline__ void atomicMaxF(float* addr, float val) {
  if (val >= 0.0f) atomicMax((int*)addr, __float_as_int(val));
  else             atomicMin((unsigned int*)addr, __float_as_uint(val));
}

__global__ void k_edge_max(const int* __restrict__ ei, int E, int Etot,
                           const float* __restrict__ asrc, const float* __restrict__ adst,
                           float* __restrict__ emax, int H) {
  int e = blockIdx.x * blockDim.x + threadIdx.x;
  if (e >= Etot) return;
  int s, d; edge_nodes(ei, E, e, s, d);
  for (int h = 0; h < H; ++h) {
    float v = asrc[s * H + h] + adst[d * H + h];
    v = (v > 0.0f) ? v : NEG_SLOPE * v;
    atomicMaxF(&emax[d * H + h], v);
  }
}

__global__ void k_edge_p(const int* __restrict__ ei, int E, int Etot,
                         const float* __restrict__ asrc, const float* __restrict__ adst,
                         const float* __restrict__ emax, float* __restrict__ p,
                         float* __restrict__ denom, int H) {
  int e = blockIdx.x * blockDim.x + threadIdx.x;
  if (e >= Etot) return;
  int s, d; edge_nodes(ei, E, e, s, d);
  for (int h = 0; h < H; ++h) {
    float v = asrc[s * H + h] + adst[d * H + h];
    v = (v > 0.0f) ? v : NEG_SLOPE * v;
    float pv = __expf(v - emax[d * H + h]);
    p[(size_t)e * H + h] = pv;
    atomicAdd(&denom[d * H + h], pv);
  }
}

// thread per (edge, h, c); consecutive threads walk c -> coalesced gather/scatter
__global__ void k_edge_msg(const int* __restrict__ ei, int E, int Etot,
                           const float* __restrict__ xh, const float* __restrict__ p,
                           const float* __restrict__ denom, float* __restrict__ out,
                           int H, int C, int ldx, int ldo) {
  long long t = (long long)blockIdx.x * blockDim.x + threadIdx.x;
  const int HC = H * C;
  const long long tot = (long long)Etot * HC;
  if (t >= tot) return;
  int e = (int)(t / HC);
  int r = (int)(t % HC);
  int h = r / C, c = r % C;
  int s, d; edge_nodes(ei, E, e, s, d);
  float alpha = p[(size_t)e * H + h] / (denom[d * H + h] + 1e-16f);
  atomicAdd(&out[(size_t)d * ldo + r],
            xh[(size_t)s * ldx + h * C + c] * alpha);
}

__global__ void k_bias_relu(float* __restrict__ h, const float* __restrict__ bias,
                            int total, int F) {
  int i = blockIdx.x * blockDim.x + threadIdx.x;
  if (i >= total) return;
  float v = h[i] + bias[i % F];
  h[i] = (v > 0.0f) ? v : 0.0f;
}

__global__ void k_bias_sigmoid(float* __restrict__ o, const float* __restrict__ bias,
                               int total, int F) {
  int i = blockIdx.x * blockDim.x + threadIdx.x;
  if (i >= total) return;
  float v = o[i] + bias[i % F];
  o[i] = 1.0f / (1.0f + __expf(-v));
}

extern "C" void kernel_launch(void* const* d_in, const int* in_sizes, int n_in,
                              void* d_out, int out_size, void* d_ws, size_t ws_size,
                              hipStream_t stream) {
  (void)n_in; (void)out_size; (void)ws_size;
  const float* x   = (const float*)d_in[0];
  const int*   ei  = (const int*)d_in[1];
  const float* W1  = (const float*)d_in[2];
  const float* as1 = (const float*)d_in[3];
  const float* ad1 = (const float*)d_in[4];
  const float* b1  = (const float*)d_in[5];
  const float* W2  = (const float*)d_in[6];
  const float* as2 = (const float*)d_in[7];
  const float* ad2 = (const float*)d_in[8];
  const float* b2  = (const float*)d_in[9];

  const int E    = in_sizes[1] / 2;
  const int Etot = E + NNODES;

  // workspace carve-out (floats), ~65 MB total
  float* w = (float*)d_ws;
  size_t off = 0;
  auto carve = [&](size_t nElem) { float* r = w + off; off += nElem; return r; };
  float* xpad  = carve((size_t)NNODES * KP1);   // x padded 50 -> 52 cols
  float* Wt1   = carve((size_t)F1 * KP1);       // W1^T  [256 x 52]
  float* Wt2   = carve((size_t)LD2 * F1);       // W2^T  [128 x 256] (N padded)
  float* xh1   = carve((size_t)NNODES * F1);
  float* av1s  = carve((size_t)NNODES * H1);
  float* av1d  = carve((size_t)NNODES * H1);
  float* emax1 = carve((size_t)NNODES * H1);
  float* den1  = carve((size_t)NNODES * H1);
  float* p1    = carve((size_t)Etot * H1);
  float* hbuf  = carve((size_t)NNODES * F1);
  float* xh2   = carve((size_t)NNODES * LD2);
  float* av2s  = carve((size_t)NNODES);
  float* av2d  = carve((size_t)NNODES);
  float* emax2 = carve((size_t)NNODES);
  float* den2  = carve((size_t)NNODES);
  float* p2    = carve((size_t)Etot);
  float* outf  = (float*)d_out;

  const int TB = 256;
  auto cdiv = [](long long a, long long b) { return (int)((a + b - 1) / b); };

  // ---------------- Layer 1 ----------------
  k_pad<<<cdiv((long long)NNODES * KP1, TB), TB, 0, stream>>>(x, xpad, NNODES, F_IN, KP1);
  k_padT<<<cdiv((long long)F1 * KP1, TB), TB, 0, stream>>>(W1, Wt1, F_IN, F1, KP1, F1);
  // xh1 = x @ W1 via fp32 WMMA (both fragments: b64 loads at immediate offsets)
  k_gemm_wmma<KP1><<<dim3(NNODES / 16, F1 / 64), 128, 0, stream>>>(xpad, Wt1, xh1, F1);
  k_att<<<cdiv(NNODES * H1, TB), TB, 0, stream>>>(xh1, as1, av1s, NNODES * H1, H1, C1, F1);
  k_att<<<cdiv(NNODES * H1, TB), TB, 0, stream>>>(xh1, ad1, av1d, NNODES * H1, H1, C1, F1);
  k_fill<<<cdiv(NNODES * H1, TB), TB, 0, stream>>>(emax1, -INFINITY, NNODES * H1);
  k_fill<<<cdiv(NNODES * H1, TB), TB, 0, stream>>>(den1, 0.0f, NNODES * H1);
  k_fill<<<cdiv((long long)NNODES * F1, TB), TB, 0, stream>>>(hbuf, 0.0f, NNODES * F1);
  k_edge_max<<<cdiv(Etot, TB), TB, 0, stream>>>(ei, E, Etot, av1s, av1d, emax1, H1);
  k_edge_p<<<cdiv(Etot, TB), TB, 0, stream>>>(ei, E, Etot, av1s, av1d, emax1, p1, den1, H1);
  k_edge_msg<<<cdiv((long long)Etot * F1, TB), TB, 0, stream>>>(
      ei, E, Etot, xh1, p1, den1, hbuf, H1, C1, F1, F1);
  k_bias_relu<<<cdiv((long long)NNODES * F1, TB), TB, 0, stream>>>(hbuf, b1, NNODES * F1, F1);

  // ---------------- Layer 2 ----------------
  k_padT<<<cdiv((long long)LD2 * F1, TB), TB, 0, stream>>>(W2, Wt2, F1, NCLS, F1, LD2);
  // xh2 = hbuf @ W2 (K=256, N padded to 128 -> no guards)
  k_gemm_wmma<F1><<<dim3(NNODES / 16, LD2 / 64), 128, 0, stream>>>(hbuf, Wt2, xh2, LD2);
  k_att<<<cdiv(NNODES, TB), TB, 0, stream>>>(xh2, as2, av2s, NNODES, 1, NCLS, LD2);
  k_att<<<cdiv(NNODES, TB), TB, 0, stream>>>(xh2, ad2, av2d, NNODES, 1, NCLS, LD2);
  k_fill<<<cdiv(NNODES, TB), TB, 0, stream>>>(emax2, -INFINITY, NNODES);
  k_fill<<<cdiv(NNODES, TB), TB, 0, stream>>>(den2, 0.0f, NNODES);
  k_fill<<<cdiv((long long)NNODES * NCLS, TB), TB, 0, stream>>>(outf, 0.0f, NNODES * NCLS);
  k_edge_max<<<cdiv(Etot, TB), TB, 0, stream>>>(ei, E, Etot, av2s, av2d, emax2, 1);
  k_edge_p<<<cdiv(Etot, TB), TB, 0, stream>>>(ei, E, Etot, av2s, av2d, emax2, p2, den2, 1);
  k_edge_msg<<<cdiv((long long)Etot * NCLS, TB), TB, 0, stream>>>(
      ei, E, Etot, xh2, p2, den2, outf, 1, NCLS, LD2, NCLS);
  k_bias_sigmoid<<<cdiv((long long)NNODES * NCLS, TB), TB, 0, stream>>>(
      outf, b2, NNODES * NCLS, NCLS);
}